// FocalLoss_42468636623585
// MI455X (gfx1250) — compile-verified
//
#include <hip/hip_runtime.h>

typedef __attribute__((ext_vector_type(2))) float v2f;
typedef __attribute__((ext_vector_type(8))) float v8f;

#define NUM_CLASSES 4
#define NUM_M 64
#define WAVES_PER_BLOCK 8
#define GROUPS_PER_WAVE 8                      // 16-anchor groups per wave
#define ANCHORS_PER_BLOCK (WAVES_PER_BLOCK * GROUPS_PER_WAVE * 16)  // 1024

#define EPSV 1e-4f
#define POS_D2P1 401.0f    /* 20^2 + 1 (distances biased by +1 via WMMA C) */
#define BG_D2P1  901.0f    /* 30^2 + 1 */
#define POS_ANG 15.0f
#define BG_ANG  22.5f

__device__ __forceinline__ unsigned umin2(unsigned a, unsigned b) { return a < b ? a : b; }

__global__ void fl_zero_ws(float* ws, int n) {
    int i = blockIdx.x * blockDim.x + threadIdx.x;
    if (i < n) ws[i] = 0.0f;
}

__global__ __launch_bounds__(256) void fl_main(
    const float* __restrict__ cls,   // (B, A, 4)
    const float* __restrict__ reg,   // (B, A, 3)
    const float* __restrict__ anc,   // (A, 3)
    const float* __restrict__ ann,   // (B, 64, 4)
    float* __restrict__ ws,          // (B, 4): cls_sum, xy_sum, ang_sum, n_pos
    int A)
{
    __shared__ float s_t0[NUM_M], s_t1[NUM_M], s_t2[NUM_M];
    __shared__ float s_mx[NUM_M], s_my[NUM_M], s_mal[NUM_M], s_mlab[NUM_M];

    const int b   = blockIdx.y;
    const int tid = threadIdx.x;

    // Preload this sample's 64 annotations; precompute WMMA A-matrix terms.
    if (tid < NUM_M) {
        const float4 a4 = ((const float4*)ann)[b * NUM_M + tid];
        s_mx[tid]   = a4.x;
        s_my[tid]   = a4.y;
        s_mal[tid]  = a4.z;
        s_mlab[tid] = a4.w;
        s_t0[tid]   = -2.0f * a4.x;
        s_t1[tid]   = -2.0f * a4.y;
        // invalid annotation -> entire D row = +inf (matches jnp.where(valid, d, inf))
        s_t2[tid]   = (a4.w != -1.0f) ? (a4.x * a4.x + a4.y * a4.y) : __builtin_inff();
    }
    __syncthreads();

    const int      wave  = tid >> 5;
    const int      lane  = tid & 31;
    const int      col   = lane & 15;
    const bool     hi    = lane >= 16;
    const unsigned hibit = hi ? 32u : 0u;   // lane-half row offset flag (bit 5 of key)

    // A fragments (16 annotations x K=4), fixed for the whole block.
    // ISA layout: lanes 0-15 hold K=0,1 ; lanes 16-31 hold K=2,3 for rows M=0..15.
    v2f afrag[4];
#pragma unroll
    for (int t = 0; t < 4; ++t) {
        const int r = t * 16 + col;
        if (!hi) { afrag[t].x = s_t0[r]; afrag[t].y = s_t1[r]; }
        else     { afrag[t].x = s_t2[r]; afrag[t].y = 1.0f;    }
    }

    // Accumulator of ones: D = A*B + 1 biases all squared distances positive
    // (raw d2 can be ~-0.03 from cancellation), making IEEE bits sortable as u32.
    v8f cone;
#pragma unroll
    for (int i = 0; i < 8; ++i) cone[i] = 1.0f;

    float p_cls = 0.0f, p_xy = 0.0f, p_ang = 0.0f, p_pos = 0.0f;

    const int waveBase = blockIdx.x * ANCHORS_PER_BLOCK + wave * (GROUPS_PER_WAVE * 16);

    for (int it = 0; it < GROUPS_PER_WAVE; ++it) {
        const int aIdx = min(waveBase + it * 16 + col, A - 1);
        const float ax = anc[aIdx * 3 + 0];
        const float ay = anc[aIdx * 3 + 1];

        // B fragment (K=4 x 16 anchors): lanes 0-15 rows K=0,1 ; lanes 16-31 rows K=2,3.
        v2f bfrag;
        if (!hi) { bfrag.x = ax;   bfrag.y = ay; }
        else     { bfrag.x = 1.0f; bfrag.y = ax * ax + ay * ay; }

        // Packed-key argmin: key = float_bits(d2+1) | slot, slot = t*8+v (immediate).
        unsigned best = 0xFFFFFFFFu;
#pragma unroll
        for (int t = 0; t < 4; ++t) {
            // D[m][n] = ||m||^2 + ||a_n||^2 - 2 m.a_n + 1  (= squared distance + 1)
            v8f d = __builtin_amdgcn_wmma_f32_16x16x4_f32(
                false, afrag[t], false, bfrag, (short)0, cone, false, false);
            unsigned k[8];
#pragma unroll
            for (int v = 0; v < 8; ++v)
                k[v] = __float_as_uint(d[v]) | (unsigned)(t * 8 + v);
            const unsigned t0 = umin2(umin2(k[0], k[1]), umin2(k[2], k[3]));
            const unsigned t1 = umin2(umin2(k[4], k[5]), umin2(k[6], k[7]));
            best = umin2(best, umin2(t0, t1));
        }
        best |= hibit;  // tag lane-half once, not per element

        // Merge lane pair (l, l^16): together they cover all 16 rows of the column.
        best = umin2(best, (unsigned)__shfl_xor((int)best, 16, 32));

        if (!hi) {  // lanes 0-15 own one anchor each for the scalar tail
            const int s = (int)(best & 31u);
            const int bestm = ((s >> 3) << 4) + (s & 7) + (int)((best >> 2) & 8u); // t*16+v+hi*8
            const float dp1 = __uint_as_float(best & 0xFFFFFFC0u);  // d2+1 (low bits masked)

            const float aal  = anc[aIdx * 3 + 2];
            const float mx   = s_mx[bestm];
            const float my   = s_my[bestm];
            const float mal  = s_mal[bestm];
            const float mlab = s_mlab[bestm];

            const float adiff      = fabsf(aal - mal);
            const bool  positive   = (dp1 < POS_D2P1) && (adiff < POS_ANG);
            const bool  background = (dp1 >= BG_D2P1) || (adiff >= BG_ANG);

            if (positive || background) {
                const float4 c4 = ((const float4*)cls)[(size_t)b * A + aIdx];
                float p[4] = {c4.x, c4.y, c4.z, c4.w};
                int label = (int)mlab;
                label = label < 0 ? 0 : (label > 3 ? 3 : label);
#pragma unroll
                for (int cc = 0; cc < 4; ++cc) {
                    float pv = fminf(fmaxf(p[cc], EPSV), 1.0f - EPSV);
                    if (positive && cc == label) {
                        const float q = 1.0f - pv;
                        p_cls += 0.95f * q * q * (-__logf(pv));
                    } else {
                        p_cls += 0.05f * pv * pv * (-__logf(1.0f - pv));
                    }
                }
            }
            if (positive) {
                const size_t rbase = ((size_t)b * A + aIdx) * 3;
                const float rx = reg[rbase + 0];
                const float ry = reg[rbase + 1];
                const float ra = reg[rbase + 2];
                const float tx = mx - ax, ty = my - ay, ta = mal - aal;
                const float d0 = fabsf(tx - rx);
                const float d1 = fabsf(ty - ry);
                p_xy += (d0 <= (1.0f / 9.0f)) ? 4.5f * d0 * d0 : d0 - (0.5f / 9.0f);
                p_xy += (d1 <= (1.0f / 9.0f)) ? 4.5f * d1 * d1 : d1 - (0.5f / 9.0f);
                p_ang += fmaxf((fabsf(ta - ra) - 10.0f) * 0.2f, 0.0f);
                p_pos += 1.0f;
            }
        }
    }

    // Wave32 reduction, then one atomic per value per wave.
#pragma unroll
    for (int off = 16; off > 0; off >>= 1) {
        p_cls += __shfl_down(p_cls, off, 32);
        p_xy  += __shfl_down(p_xy,  off, 32);
        p_ang += __shfl_down(p_ang, off, 32);
        p_pos += __shfl_down(p_pos, off, 32);
    }
    if (lane == 0) {
        atomicAdd(&ws[b * 4 + 0], p_cls);
        atomicAdd(&ws[b * 4 + 1], p_xy);
        atomicAdd(&ws[b * 4 + 2], p_ang);
        atomicAdd(&ws[b * 4 + 3], p_pos);
    }
}

__global__ void fl_finalize(const float* __restrict__ ws, float* __restrict__ out, int B) {
    if (threadIdx.x == 0 && blockIdx.x == 0) {
        float sc = 0.0f, sx = 0.0f, sa = 0.0f;
        for (int bb = 0; bb < B; ++bb) {
            const float denom = fmaxf(ws[bb * 4 + 3], 1.0f);
            sc += ws[bb * 4 + 0] / denom;
            sx += ws[bb * 4 + 1] / (2.0f * denom);
            sa += ws[bb * 4 + 2] / denom;
        }
        const float invB = 1.0f / (float)B;
        out[0] = sc * invB;
        out[1] = sx * invB;
        out[2] = sa * invB;
    }
}

extern "C" void kernel_launch(void* const* d_in, const int* in_sizes, int n_in,
                              void* d_out, int out_size, void* d_ws, size_t ws_size,
                              hipStream_t stream) {
    (void)n_in; (void)out_size; (void)ws_size;
    const float* cls = (const float*)d_in[0];  // (B, A, 4)
    const float* reg = (const float*)d_in[1];  // (B, A, 3)
    const float* anc = (const float*)d_in[2];  // (1, A, 3)
    const float* ann = (const float*)d_in[3];  // (B, 64, 4)

    const int A = in_sizes[2] / 3;
    const int B = in_sizes[3] / (NUM_M * 4);

    float* ws = (float*)d_ws;

    fl_zero_ws<<<1, 64, 0, stream>>>(ws, B * 4);

    dim3 grid((A + ANCHORS_PER_BLOCK - 1) / ANCHORS_PER_BLOCK, B);
    fl_main<<<grid, 256, 0, stream>>>(cls, reg, anc, ann, ws, A);

    fl_finalize<<<1, 32, 0, stream>>>(ws, (float*)d_out, B);
}